// DropoutConv2d_54047868452972
// MI455X (gfx1250) — compile-verified
//
#include <hip/hip_runtime.h>

typedef float v2f __attribute__((ext_vector_type(2)));
typedef float v4f __attribute__((ext_vector_type(4)));
typedef float v8f __attribute__((ext_vector_type(8)));

#define ROWS    4        // output rows per workgroup
#define TILE_W  128      // output x-positions per workgroup (8 waves x 16)
#define KPAD    156      // 150 real taps padded; 156 = 4*39 keeps B loads bank-conflict-free
#define XROW    136      // LDS row stride for input tile (132 cols used)
#define CH_ROWS 8        // input rows staged per channel (ROWS + 4 halo)
#define NROWS   48       // 6 channels * 8 rows
#define OSTR    132      // out-staging stride (breaks 128-stride bank conflict)

// LDS offset of tap k (k = ci*25 + kh*5 + kw); rows are ci*8 + kh
#define GK(k) ((((k) / 25) * CH_ROWS + ((k) % 25) / 5) * XROW + ((k) % 5))

__global__ __launch_bounds__(256)
void c3_conv_wmma_f32(const float* __restrict__ x, const float* __restrict__ W,
                      const float* __restrict__ bias, const int* __restrict__ mask,
                      float* __restrict__ out) {
  __shared__ __align__(16) float sW[16 * KPAD];   // masked weights [n][k]
  __shared__ __align__(16) float sX[ROWS * 16 * OSTR]; // 8448 >= 52*136; tile then out-staging

  const int tid = threadIdx.x;
  const int x0  = blockIdx.x * TILE_W;
  const int y   = blockIdx.y * ROWS;
  const int b   = blockIdx.z;

  // ---- stage masked weights: sW[n*KPAD + k], k = ci*25+kh*5+kw (coalesced W reads) ----
  for (int j = tid; j < 2400; j += 256) {
    int n  = j / 150;
    int k  = j - n * 150;
    int ci = k / 25;
    sW[n * KPAD + k] = W[j] * (float)mask[n * 6 + ci];
  }
  if (tid < 128 && (tid & 7) < 6)                  // zero K-pad taps 150..155
    sW[(tid >> 3) * KPAD + 150 + (tid & 7)] = 0.0f;

  // ---- stage input tile: 6 ch x 8 rows x 132 cols, stride 136; rows 48..51 zeroed ----
  {
    const float* xb = x + (size_t)b * (6 * 512 * 512);
    const int col   = tid & 127;
    const int rsel  = tid >> 7;                    // 0/1: which of the 2 rows per sweep
    const int ybase = (y - 2) * 512 + (x0 + col - 2);
    const bool gxok = (unsigned)(x0 + col - 2) < 512u;
    const int stb   = rsel * XROW + col;
    #pragma unroll
    for (int it = 0; it < 24; ++it) {              // 48 rows, 2 per sweep, cols 0..127
      const int rowA = 2 * it, rowB = 2 * it + 1;  // compile-time
      const int offA = (rowA >> 3) * (512 * 512) + (rowA & 7) * 512;
      const int offB = (rowB >> 3) * (512 * 512) + (rowB & 7) * 512;
      int off = rsel ? offB : offA;
      int rr  = rsel ? (rowB & 7) : (rowA & 7);
      int gy  = y + rr - 2;
      float v = 0.0f;
      if (gxok && (unsigned)gy < 512u) v = xb[off + ybase];
      sX[it * (2 * XROW) + stb] = v;
    }
    if (tid < 192) {                               // tail cols 128..131, all 48 rows
      int row = tid >> 2;
      int c2  = 128 + (tid & 3);
      int gy  = y + (row & 7) - 2;
      int gx  = x0 + c2 - 2;
      float v = 0.0f;
      if ((unsigned)gy < 512u && (unsigned)gx < 512u)
        v = xb[(row >> 3) * (512 * 512) + gy * 512 + gx];
      sX[row * XROW + c2] = v;
    }
    for (int i = tid; i < 4 * XROW; i += 256)      // zero rows 48..51 (K-pad reads)
      sX[NROWS * XROW + i] = 0.0f;
  }
  __syncthreads();

  const int lane = tid & 31;
  const int w    = tid >> 5;
  const int mcol = lane & 15;          // A: M index / B: N index
  const int kg   = (lane >> 4) << 1;   // K sub-offset 0 / 2 per A,B VGPR layout
  const int lanebase = w * 16 + mcol;

  // g(k+2)-g(k) in {2,133,541}: three per-lane bases; every other offset is a DS immediate
  const float* aB2   = sX + lanebase + (kg ? 2   : 0);
  const float* aB133 = sX + lanebase + (kg ? 133 : 0);
  const float* aB541 = sX + lanebase + (kg ? 541 : 0);
  const float* bBase = sW + mcol * KPAD + kg;      // even index -> 8B-aligned b64

  v8f ac0 = {}, ac1 = {}, ac2 = {}, ac3 = {};
  #pragma unroll
  for (int c = 0; c < 38; ++c) {                   // K = 152 = 38 chunks of 4
    const int k0 = 4 * c;                          // compile-time after unroll
    const int f0 = GK(k0), f1 = GK(k0 + 1);
    const int d0 = GK(k0 + 2) - f0;
    const int d1 = GK(k0 + 3) - f1;
    const float* p0 = (d0 == 2) ? aB2 : (d0 == 133) ? aB133 : aB541;
    const float* p1 = (d1 == 2) ? aB2 : (d1 == 133) ? aB133 : aB541;
    v2f bv = *(const v2f*)(bBase + k0);            // shared by all 4 row-WMMAs
    v2f a0; a0.x = p0[f0           ]; a0.y = p1[f1           ];
    v2f a1; a1.x = p0[f0 +     XROW]; a1.y = p1[f1 +     XROW];
    v2f a2; a2.x = p0[f0 + 2 * XROW]; a2.y = p1[f1 + 2 * XROW];
    v2f a3; a3.x = p0[f0 + 3 * XROW]; a3.y = p1[f1 + 3 * XROW];
    ac0 = __builtin_amdgcn_wmma_f32_16x16x4_f32(false, a0, false, bv, (short)0, ac0, false, false);
    ac1 = __builtin_amdgcn_wmma_f32_16x16x4_f32(false, a1, false, bv, (short)0, ac1, false, false);
    ac2 = __builtin_amdgcn_wmma_f32_16x16x4_f32(false, a2, false, bv, (short)0, ac2, false, false);
    ac3 = __builtin_amdgcn_wmma_f32_16x16x4_f32(false, a3, false, bv, (short)0, ac3, false, false);
  }

  __syncthreads();                     // all waves done reading the input tile
  {                                    // transpose 16x16 tiles through LDS
    const int mbase = (lane >> 4) << 3;          // C/D: lanes 16-31 hold M = r + 8
    #pragma unroll
    for (int r = 0; r < 8; ++r) {
      const int cidx = w * 16 + mbase + r;
      sX[(0 * 16 + mcol) * OSTR + cidx] = ac0[r];
      sX[(1 * 16 + mcol) * OSTR + cidx] = ac1[r];
      sX[(2 * 16 + mcol) * OSTR + cidx] = ac2[r];
      sX[(3 * 16 + mcol) * OSTR + cidx] = ac3[r];
    }
  }
  __syncthreads();

  // ---- b128 non-temporal coalesced writeout, bias fused ----
  for (int q = tid; q < 2048; q += 256) {          // 4 rows * 16 ch * 32 float4
    const int j  = q >> 9;
    const int n  = (q >> 5) & 15;
    const int xq = (q & 31) << 2;
    v4f v = *(const v4f*)(sX + (j * 16 + n) * OSTR + xq);
    const float bb = bias[n];
    v.x += bb; v.y += bb; v.z += bb; v.w += bb;
    __builtin_nontemporal_store(
        v, (v4f*)(out + (((size_t)((b * 16 + n) * 512 + (y + j))) << 9) + x0 + xq));
  }
}

extern "C" void kernel_launch(void* const* d_in, const int* in_sizes, int n_in,
                              void* d_out, int out_size, void* d_ws, size_t ws_size,
                              hipStream_t stream) {
  (void)in_sizes; (void)n_in; (void)d_ws; (void)ws_size; (void)out_size;
  const float* x    = (const float*)d_in[0];
  const float* W    = (const float*)d_in[1];
  const float* bias = (const float*)d_in[2];
  const int*   mask = (const int*)d_in[3];
  float* out = (float*)d_out;

  dim3 grid(512 / TILE_W, 512 / ROWS, 32);   // x-tiles, row-tiles, batch
  c3_conv_wmma_f32<<<grid, 256, 0, stream>>>(x, W, bias, mask, out);
}